// GraphSAGE_12893491822859
// MI455X (gfx1250) — compile-verified
//
#include <hip/hip_runtime.h>
#include <hip/hip_bf16.h>
#include <stddef.h>

typedef float v2f __attribute__((ext_vector_type(2)));
typedef float v8f __attribute__((ext_vector_type(8)));

#define D_FEAT 128

// ---------------------------------------------------------------------------
// Kernel 1: zero a float buffer (harness poisons d_out / d_ws with 0xAA).
// ---------------------------------------------------------------------------
__global__ void sage_zero_f32(float* __restrict__ p, int n4) {
    int i = blockIdx.x * blockDim.x + threadIdx.x;
    if (i < n4) {
        ((float4*)p)[i] = make_float4(0.f, 0.f, 0.f, 0.f);
    }
}

// ---------------------------------------------------------------------------
// Kernel 2: edge scatter. One wave32 per edge.
//   lane L holds feat[src][4L..4L+3] (float4) -> 4x global_atomic_add_f32 into
//   hsum[dst] row. lane 0 accumulates degree.
// ---------------------------------------------------------------------------
__global__ void sage_scatter(const float* __restrict__ feat,
                             const int*   __restrict__ src,
                             const int*   __restrict__ dst,
                             float*       __restrict__ hsum,   // = d_out, N x 128
                             float*       __restrict__ deg,    // N
                             int E) {
    int edge = blockIdx.x * (blockDim.x >> 5) + (threadIdx.x >> 5);
    int lane = threadIdx.x & 31;
    if (edge >= E) return;

    int s = src[edge];
    int d = dst[edge];

    const float4 v = *(const float4*)(feat + (size_t)s * D_FEAT + lane * 4);
    float* o = hsum + (size_t)d * D_FEAT + lane * 4;
    atomicAdd(o + 0, v.x);
    atomicAdd(o + 1, v.y);
    atomicAdd(o + 2, v.z);
    atomicAdd(o + 3, v.w);
    if (lane == 0) atomicAdd(deg + d, 1.0f);
}

// ---------------------------------------------------------------------------
// Kernel 3: fused in-place GEMM with fp32 WMMA (V_WMMA_F32_16X16X4_F32).
//   block = 256 threads (8 waves) owns 16 rows (one M tile).
//   Stage X tile and scaled H tile (hsum/max(deg,1)) in LDS, barrier, then
//   wave w computes output columns [16w, 16w+16) and overwrites d_out rows.
//
// A-fragment (16x4 f32, ISA 7.12.2): lane L<16: M=L, VGPR0=K0, VGPR1=K1;
//                                    lane L>=16: M=L-16, VGPR0=K2, VGPR1=K3.
// B-fragment (4x16): lane L<16: N=L, VGPR0=B[K0][N], VGPR1=B[K1][N];
//                    lane L>=16: N=L-16, K2/K3.   (mirror of A)
// C/D (16x16 f32): VGPR r, lane L: M = r + (L>=16 ? 8 : 0), N = L&15.
// ---------------------------------------------------------------------------
__global__ void __launch_bounds__(256)
sage_gemm(const float* __restrict__ feat,
          const float* __restrict__ Wself,    // [128][128] row-major
          const float* __restrict__ Wneigh,   // [128][128] row-major
          const float* __restrict__ bias,     // [128]
          const float* __restrict__ deg,      // [N]
          float*       __restrict__ out)      // in: hsum, out: result (in place)
{
    __shared__ float tX[16 * D_FEAT];   // X tile, 8 KB
    __shared__ float tH[16 * D_FEAT];   // scaled neighbor-mean tile, 8 KB

    const int mBase = blockIdx.x * 16;
    const int tid   = threadIdx.x;

    // --- stage tiles: 16 threads per row, 8 floats (2 x float4) per thread ---
    {
        const int row = tid >> 4;            // 0..15
        const int col = (tid & 15) * 8;      // 0,8,..,120
        const size_t g = (size_t)(mBase + row) * D_FEAT + col;

        float s = deg[mBase + row];
        s = 1.0f / fmaxf(s, 1.0f);

        float4 x0 = *(const float4*)(feat + g);
        float4 x1 = *(const float4*)(feat + g + 4);
        float4 h0 = *(const float4*)(out + g);
        float4 h1 = *(const float4*)(out + g + 4);
        h0.x *= s; h0.y *= s; h0.z *= s; h0.w *= s;
        h1.x *= s; h1.y *= s; h1.z *= s; h1.w *= s;

        *(float4*)(tX + row * D_FEAT + col)     = x0;
        *(float4*)(tX + row * D_FEAT + col + 4) = x1;
        *(float4*)(tH + row * D_FEAT + col)     = h0;
        *(float4*)(tH + row * D_FEAT + col + 4) = h1;
    }
    __syncthreads();

    // --- WMMA: each wave owns one 16-wide N tile ---
    const int wave = tid >> 5;           // 0..7
    const int lane = tid & 31;
    const int n    = wave * 16;          // output column base
    const int ml   = lane & 15;          // M (A / C lanes) and N offset (B lanes)
    const int koff = (lane >> 4) * 2;    // 0 for lanes 0-15, 2 for lanes 16-31

    v8f acc1 = {};  // X @ Wself
    v8f acc2 = {};  // h @ Wneigh

    #pragma unroll 4
    for (int kk = 0; kk < D_FEAT; kk += 4) {
        // A fragments: contiguous K pair -> 8-byte LDS load
        v2f ax = *(const v2f*)(tX + ml * D_FEAT + kk + koff);
        v2f ah = *(const v2f*)(tH + ml * D_FEAT + kk + koff);

        // B fragments: column n+ml of W, rows kk+koff, kk+koff+1
        v2f bs, bn;
        bs.x = Wself [(size_t)(kk + koff)     * D_FEAT + n + ml];
        bs.y = Wself [(size_t)(kk + koff + 1) * D_FEAT + n + ml];
        bn.x = Wneigh[(size_t)(kk + koff)     * D_FEAT + n + ml];
        bn.y = Wneigh[(size_t)(kk + koff + 1) * D_FEAT + n + ml];

        acc1 = __builtin_amdgcn_wmma_f32_16x16x4_f32(
            false, ax, false, bs, (short)0, acc1, false, false);
        acc2 = __builtin_amdgcn_wmma_f32_16x16x4_f32(
            false, ah, false, bn, (short)0, acc2, false, false);
    }

    // --- epilogue: out = acc1 + acc2 + b, overwrite own rows in place ---
    const float bv   = bias[n + ml];
    const int   mrow = mBase + (lane >> 4) * 8;   // + r per accumulator VGPR
    #pragma unroll
    for (int r = 0; r < 8; ++r) {
        out[(size_t)(mrow + r) * D_FEAT + n + ml] = acc1[r] + acc2[r] + bv;
    }
}

// ---------------------------------------------------------------------------
// Host launcher. Inputs: features[N*128], src[E], dst[E],
//                        W_self[128*128], W_neigh[128*128], b[128].
// d_out: N*128 fp32. d_ws: first N floats = degree accumulator.
// ---------------------------------------------------------------------------
extern "C" void kernel_launch(void* const* d_in, const int* in_sizes, int n_in,
                              void* d_out, int out_size, void* d_ws, size_t ws_size,
                              hipStream_t stream) {
    const float* feat   = (const float*)d_in[0];
    const int*   src    = (const int*)d_in[1];
    const int*   dst    = (const int*)d_in[2];
    const float* Wself  = (const float*)d_in[3];
    const float* Wneigh = (const float*)d_in[4];
    const float* bias   = (const float*)d_in[5];

    const int N = in_sizes[0] / D_FEAT;   // 50000
    const int E = in_sizes[1];            // 800000

    float* out = (float*)d_out;
    float* deg = (float*)d_ws;            // N floats

    // 1) zero the sum accumulator (d_out) and degrees (d_ws)
    {
        int n4 = (N * D_FEAT) / 4;
        sage_zero_f32<<<(n4 + 255) / 256, 256, 0, stream>>>(out, n4);
        int d4 = (N + 3) / 4;             // deg buffer zeroed in float4 chunks
        sage_zero_f32<<<(d4 + 255) / 256, 256, 0, stream>>>(deg, d4);
    }

    // 2) edge scatter: 8 edges per 256-thread block (one wave per edge)
    {
        int edgesPerBlock = 256 / 32;
        int blocks = (E + edgesPerBlock - 1) / edgesPerBlock;
        sage_scatter<<<blocks, 256, 0, stream>>>(feat, src, dst, out, deg, E);
    }

    // 3) fused WMMA GEMM, in place over d_out
    {
        int blocks = N / 16;              // 3125 (N divisible by 16)
        sage_gemm<<<blocks, 256, 0, stream>>>(feat, Wself, Wneigh, bias, deg, out);
    }
}